// LengthRegulator_89197880803889
// MI455X (gfx1250) — compile-verified
//
#include <hip/hip_runtime.h>

// Problem constants (reference: N=16, L=512, D=384, T=4096)
#define NB 16
#define LL 512
#define DD 384
#define TT 4096

typedef float v4f __attribute__((ext_vector_type(4)));

// ---------------------------------------------------------------------------
// Kernel A: per-batch inclusive cumsum of int64 durations -> int32 cum table.
// One 512-thread block per batch row; Hillis-Steele scan in LDS.
// ---------------------------------------------------------------------------
__global__ __launch_bounds__(LL) void lr_cumsum_kernel(
    const long long* __restrict__ dur, int* __restrict__ cum) {
  __shared__ int s[LL];
  const int n = blockIdx.x;
  const int l = threadIdx.x;
  s[l] = (int)dur[n * LL + l];
  __syncthreads();
#pragma unroll
  for (int off = 1; off < LL; off <<= 1) {
    int t = (l >= off) ? s[l - off] : 0;
    __syncthreads();
    s[l] += t;
    __syncthreads();
  }
  cum[n * LL + l] = s[l];
}

// ---------------------------------------------------------------------------
// Kernel B: gather rows of x into out according to cum, zero-pad the tail.
// 256 threads = 8 wave32s per block; each block covers 64 consecutive frames
// of one batch row. cum row (2KB) staged to LDS via CDNA5 async-to-LDS DMA.
// Each wave handles 8 frames; per frame: 9-step LDS binary search (uniform)
// then 3x b128 per lane row copy with non-temporal stores.
// ---------------------------------------------------------------------------
__global__ __launch_bounds__(256) void lr_gather_kernel(
    const float* __restrict__ x, const int* __restrict__ cum,
    float* __restrict__ out) {
  __shared__ int cumLds[LL];

  const int tid   = threadIdx.x;
  const int n     = blockIdx.x >> 6;            // 64 blocks per batch row
  const int tbase = (blockIdx.x & 63) * 64;     // first frame of this block

  // ---- Stage cum[n][0..511] (2KB) into LDS with async DMA (ASYNCcnt path).
  // Each thread moves 8 bytes: lds[tid*8 .. +7] = cumRow[tid*2 .. tid*2+1].
  {
    const int* cumRow = cum + n * LL;                        // uniform -> SGPR pair
    unsigned lds_off = (unsigned)(uintptr_t)(&cumLds[0]) + (unsigned)tid * 8u;
    unsigned g_off   = (unsigned)tid * 8u;                   // per-lane voffset
    asm volatile("global_load_async_to_lds_b64 %0, %1, %2 offset:0"
                 :: "v"(lds_off), "v"(g_off), "s"(cumRow)
                 : "memory");
#if __has_builtin(__builtin_amdgcn_s_wait_asynccnt)
    __builtin_amdgcn_s_wait_asynccnt(0);
#else
    asm volatile("s_wait_asynccnt 0" ::: "memory");
#endif
  }
  __syncthreads();

  const int wave = tid >> 5;
  const int lane = tid & 31;
  const int total = cumLds[LL - 1];

  const v4f* __restrict__ xrows = (const v4f*)x;  // row stride = 96 v4f
  v4f* __restrict__ orows       = (v4f*)out;

#pragma unroll
  for (int i = 0; i < 8; ++i) {
    const int t = tbase + wave * 8 + i;           // this wave's frame
    v4f* dst = orows + (size_t)(n * TT + t) * 96 + lane;

    if (t < total) {
      // idx = searchsorted(cum, t, side='right'): first j with cum[j] > t.
      int lo = 0, hi = LL;
#pragma unroll
      for (int step = 0; step < 9; ++step) {      // log2(512) = 9 iterations
        int mid = (lo + hi) >> 1;
        if (cumLds[mid] <= t) lo = mid + 1; else hi = mid;
      }
      const int idx = lo < LL ? lo : LL - 1;
      const v4f* src = xrows + (size_t)(n * LL + idx) * 96 + lane;
#pragma unroll
      for (int k = 0; k < 3; ++k) {
        v4f v = src[k * 32];
        __builtin_nontemporal_store(v, dst + k * 32);
      }
    } else {
      v4f z = {0.f, 0.f, 0.f, 0.f};
#pragma unroll
      for (int k = 0; k < 3; ++k) {
        __builtin_nontemporal_store(z, dst + k * 32);
      }
    }
  }
}

// ---------------------------------------------------------------------------
extern "C" void kernel_launch(void* const* d_in, const int* in_sizes, int n_in,
                              void* d_out, int out_size, void* d_ws, size_t ws_size,
                              hipStream_t stream) {
  const float*     x   = (const float*)d_in[0];
  const long long* dur = (const long long*)d_in[1];
  // d_in[2] = mel_max_len scalar (4096) -- baked in as TT.

  int* cum = (int*)d_ws;  // N*L int32 = 32 KB scratch

  lr_cumsum_kernel<<<NB, LL, 0, stream>>>(dur, cum);
  lr_gather_kernel<<<NB * (TT / 64), 256, 0, stream>>>(x, cum, (float*)d_out);
}